// CostVolumeConcat_72602127171747
// MI455X (gfx1250) — compile-verified
//
#include <hip/hip_runtime.h>

#define B_ 4
#define C_ 32
#define H_ 64
#define W_ 128
#define D_ 48

typedef float f32x4 __attribute__((ext_vector_type(4)));
typedef int   v4i   __attribute__((vector_size(16)));          // matches builtin param type
typedef v4i   lds_v4i __attribute__((address_space(3)));       // AS3-qualified vector pointee

__global__ __launch_bounds__(256) void CostVolumeConcat_kernel(
    const float* __restrict__ left,
    const float* __restrict__ right,
    float* __restrict__ out)
{
    // LDS: [0..127] = left row, [128..255] = right row (1 KB)
    __shared__ __align__(16) float rows[2 * W_];

    const int tid = threadIdx.x;
    const int bch = blockIdx.x;              // b*C*H + c*H + h
    const int h = bch & (H_ - 1);
    const int c = (bch >> 6) & (C_ - 1);
    const int b = bch >> 11;

    const int lbase = ((b * C_ + c) * H_ + h) * W_;  // same row offset for left & right

    // ---- Stage both input rows into LDS (CDNA5 async-to-LDS path) ----
#if __has_builtin(__builtin_amdgcn_global_load_async_to_lds_b128)
    if (tid < 64) {
        const float* g = (tid < 32) ? (left  + lbase + tid * 4)
                                    : (right + lbase + (tid - 32) * 4);
        __builtin_amdgcn_global_load_async_to_lds_b128(
            (v4i*)g,
            (lds_v4i*)&rows[tid * 4],
            /*offset=*/0, /*cpol=*/0);
    }
#if __has_builtin(__builtin_amdgcn_s_wait_asynccnt)
    __builtin_amdgcn_s_wait_asynccnt(0);
#else
    asm volatile("s_wait_asynccnt 0" ::: "memory");
#endif
#else
    // Fallback: plain cooperative load (256 threads, 256 floats)
    rows[tid] = (tid < W_) ? left[lbase + tid] : right[lbase + tid - W_];
#endif
    __syncthreads();

    const float* lrow = rows;
    const float* rrow = rows + W_;

    // Output strides: [B][2C][D][H][W]
    const int out_l = (b * (2 * C_) + c) * (D_ * H_ * W_) + h * W_;
    const int out_r = out_l + C_ * (D_ * H_ * W_);

    // 1536 float4 rows-of-4 per half; 256 threads -> 6 iterations.
    // Within a wave: d is uniform, w4 == lane -> coalesced b128 NT stores.
#pragma unroll
    for (int it = 0; it < 6; ++it) {
        const int t  = tid + it * 256;       // 0..1535
        const int d  = t >> 5;               // 0..47
        const int w0 = (t & 31) << 2;        // 0,4,...,124

        // Left half: aligned float4 from LDS, mask w < d to zero.
        f32x4 lv = *(const f32x4*)(lrow + w0);
        if (w0 + 0 < d) lv.x = 0.0f;
        if (w0 + 1 < d) lv.y = 0.0f;
        if (w0 + 2 < d) lv.z = 0.0f;
        if (w0 + 3 < d) lv.w = 0.0f;

        // Right half: shifted row; clamp index so the LDS load is always safe,
        // then mask (matches reference: clip() only matters where mask==0).
        const int i0 = w0 + 0 - d, i1 = w0 + 1 - d, i2 = w0 + 2 - d, i3 = w0 + 3 - d;
        const float r0 = rrow[i0 > 0 ? i0 : 0];
        const float r1 = rrow[i1 > 0 ? i1 : 0];
        const float r2 = rrow[i2 > 0 ? i2 : 0];
        const float r3 = rrow[i3 > 0 ? i3 : 0];
        f32x4 rv;
        rv.x = (i0 >= 0) ? r0 : 0.0f;
        rv.y = (i1 >= 0) ? r1 : 0.0f;
        rv.z = (i2 >= 0) ? r2 : 0.0f;
        rv.w = (i3 >= 0) ? r3 : 0.0f;

        const int dh = d * (H_ * W_);
        __builtin_nontemporal_store(lv, (f32x4*)(out + out_l + dh + w0));
        __builtin_nontemporal_store(rv, (f32x4*)(out + out_r + dh + w0));
    }
}

extern "C" void kernel_launch(void* const* d_in, const int* in_sizes, int n_in,
                              void* d_out, int out_size, void* d_ws, size_t ws_size,
                              hipStream_t stream) {
    (void)in_sizes; (void)n_in; (void)out_size; (void)d_ws; (void)ws_size;
    const float* left  = (const float*)d_in[0];
    const float* right = (const float*)d_in[1];
    float* out = (float*)d_out;

    const int nblocks = B_ * C_ * H_;  // 8192 blocks, one (b,c,h) row-pair each
    CostVolumeConcat_kernel<<<dim3(nblocks), dim3(256), 0, stream>>>(left, right, out);
}